// GumbelVectorQuantizer_78168404787478
// MI455X (gfx1250) — compile-verified
//
#include <hip/hip_runtime.h>

// ---------------- problem constants ----------------
#define B_    16
#define T_    1024
#define F_    512          // K of the GEMM
#define G_    2
#define V_    320
#define D_    128
#define NCODES (G_ * V_)   // 640  (N of the GEMM)
#define NTOK  (B_ * T_)    // 16384 (M of the GEMM)
#define TM    32           // tokens per block (2 WMMA M-tiles)
#define NBLK  (NTOK / TM)  // 512 blocks
#define TAU_INV 0.5f
#define EPS_  1e-7f
#define QOUT  ((size_t)NTOK * G_ * D_)   // 4,194,304 floats of quantized output

// dynamic LDS: [TM][NCODES] logits + [8][V_] partials
#define LDS_LOGITS_BYTES ((size_t)TM * NCODES * 4)
#define LDS_BYTES        (LDS_LOGITS_BYTES + (size_t)8 * V_ * 4)

typedef float v2f __attribute__((ext_vector_type(2)));
typedef float v8f __attribute__((ext_vector_type(8)));

// ---------------- kernel 0: zero the hard-argmax counters ----------------
__global__ void gvq_zero(int* __restrict__ counts, int n) {
    int i = blockIdx.x * blockDim.x + threadIdx.x;
    if (i < n) counts[i] = 0;
}

// ---------------- kernel 1: fused GEMM + per-token quantizer ----------------
// One block = 32 tokens (2 M-tiles). 8 waves; each wave computes 2 M-tiles x
// 5 N-tiles with V_WMMA_F32_16X16X4_F32 over K=512: each B operand load feeds
// two WMMAs, halving redundant w_proj L2 traffic vs a 16-token tile.
// Logits land in dynamic LDS; reduction phase consumes them in-place.
__global__ __launch_bounds__(256)
void gvq_main(const float* __restrict__ x,      // (NTOK, F)
              const float* __restrict__ w,      // (NCODES, F)
              const float* __restrict__ bias,   // (NCODES)
              const float* __restrict__ cb,     // (NCODES, D)
              const float* __restrict__ gum,    // (NTOK, NCODES)
              float* __restrict__ out,          // quantized (NTOK, G*D)
              int*   __restrict__ counts,       // (T, G, V) argmax counts
              float* __restrict__ blockpart)    // (NBLK, NCODES) softmax partials
{
    extern __shared__ char smem[];
    float (*lds_logits)[NCODES] = (float (*)[NCODES])smem;            // 80 KB
    float (*lds_part)[V_]       = (float (*)[V_])(smem + LDS_LOGITS_BYTES); // 10 KB

    const int blk  = blockIdx.x;                // tokens [blk*TM, blk*TM+TM)
    const int tid  = threadIdx.x;
    const int wave = tid >> 5;
    const int lane = tid & 31;
    const int hrow = lane & 15;                 // M (or N) index within a 16-tile
    const int koff = (lane >> 4) << 1;          // lanes 16-31 carry K+2 (ISA 32-bit A layout)
    const int mbase = blk * TM;

    // ---- GEMM phase: wave covers code tiles [wave*5, wave*5+5), both M-tiles ----
    v8f acc[2][5];
    #pragma unroll
    for (int mt = 0; mt < 2; ++mt)
        #pragma unroll
        for (int i = 0; i < 5; ++i)
            #pragma unroll
            for (int q = 0; q < 8; ++q) acc[mt][i][q] = 0.0f;

    const int nt0 = wave * 5;
    const float* xrow0 = x + (size_t)(mbase + hrow) * F_ + koff;
    const float* xrow1 = x + (size_t)(mbase + 16 + hrow) * F_ + koff;

    for (int k = 0; k < F_; k += 4) {
        const v2f a0 = *(const v2f*)(xrow0 + k);            // A: 16x4 fp32 tile operands
        const v2f a1 = *(const v2f*)(xrow1 + k);
        #pragma unroll
        for (int i = 0; i < 5; ++i) {
            const int col = (nt0 + i) * 16 + hrow;          // code index for this lane
            // B[k'][n] = w[n][k'] : per-lane 64-bit load of (k+koff, k+koff+1)
            const v2f b = *(const v2f*)(w + (size_t)col * F_ + k + koff);
            acc[0][i] = __builtin_amdgcn_wmma_f32_16x16x4_f32(
                            false, a0, false, b, (short)0, acc[0][i], false, false);
            acc[1][i] = __builtin_amdgcn_wmma_f32_16x16x4_f32(
                            false, a1, false, b, (short)0, acc[1][i], false, false);
        }
    }

    // C/D layout: VGPR r -> M = (lane>>4)*8 + r, N = lane&15. Add bias, spill to LDS.
    {
        const int mhalf = (lane >> 4) * 8;
        #pragma unroll
        for (int mt = 0; mt < 2; ++mt) {
            #pragma unroll
            for (int i = 0; i < 5; ++i) {
                const int col = (nt0 + i) * 16 + hrow;
                const float bb = bias[col];
                #pragma unroll
                for (int r = 0; r < 8; ++r)
                    lds_logits[mt * 16 + mhalf + r][col] = acc[mt][i][r] + bb;
            }
        }
    }
    __syncthreads();

    // ---- reduction phase: wave owns group g = wave/4, tokens (wave%4)*8 .. +7 ----
    const int g = wave >> 2;
    float pacc[10];                              // per-lane softmax-prob partials (v = lane+32j)
    #pragma unroll
    for (int j = 0; j < 10; ++j) pacc[j] = 0.0f;

    for (int r8 = 0; r8 < 8; ++r8) {
        const int tl = (wave & 3) * 8 + r8;
        const int n  = mbase + tl;

        float lv[10], zv[10];
        const float* gp = gum + (size_t)n * NCODES + g * V_ + lane;
        #pragma unroll
        for (int j = 0; j < 10; ++j) {
            lv[j] = lds_logits[tl][g * V_ + lane + 32 * j];
            zv[j] = (lv[j] + gp[32 * j]) * TAU_INV;          // gumbel-tempered logits
        }

        // --- plain softmax: max/argmax (tie -> lowest index, like jnp.argmax) ---
        float m = lv[0]; int am = lane;
        #pragma unroll
        for (int j = 1; j < 10; ++j)
            if (lv[j] > m) { m = lv[j]; am = lane + 32 * j; }
        #pragma unroll
        for (int off = 16; off >= 1; off >>= 1) {
            const float om = __shfl_xor(m, off, 32);
            const int   oi = __shfl_xor(am, off, 32);
            if (om > m || (om == m && oi < am)) { m = om; am = oi; }
        }
        float ev[10], s = 0.0f;
        #pragma unroll
        for (int j = 0; j < 10; ++j) { ev[j] = expf(lv[j] - m); s += ev[j]; }
        #pragma unroll
        for (int off = 16; off >= 1; off >>= 1) s += __shfl_xor(s, off, 32);
        const float inv_s = 1.0f / s;
        #pragma unroll
        for (int j = 0; j < 10; ++j) pacc[j] += ev[j] * inv_s;

        if (lane == 0) {                          // integer atomic: exact + deterministic
            const int t = n & (T_ - 1);
            atomicAdd(counts + ((size_t)t * G_ + g) * V_ + am, 1);
        }

        // --- gumbel softmax: argmax row of codebook, straight-through coefficient ---
        float m2 = zv[0]; int j2 = lane;
        #pragma unroll
        for (int j = 1; j < 10; ++j)
            if (zv[j] > m2) { m2 = zv[j]; j2 = lane + 32 * j; }
        #pragma unroll
        for (int off = 16; off >= 1; off >>= 1) {
            const float om = __shfl_xor(m2, off, 32);
            const int   oi = __shfl_xor(j2, off, 32);
            if (om > m2 || (om == m2 && oi < j2)) { m2 = om; j2 = oi; }
        }
        float s2 = 0.0f;
        #pragma unroll
        for (int j = 0; j < 10; ++j) s2 += expf(zv[j] - m2);
        #pragma unroll
        for (int off = 16; off >= 1; off >>= 1) s2 += __shfl_xor(s2, off, 32);
        const float sstar = 1.0f / s2;            // y_soft at its argmax
        const float coef  = (1.0f + sstar) - sstar; // y = y_hard + y_soft - sg(y_soft)

        const float* cbp = cb + ((size_t)g * V_ + j2) * D_;
        float* op = out + (size_t)n * (G_ * D_) + g * D_;
        #pragma unroll
        for (int d = lane; d < D_; d += 32) op[d] = coef * cbp[d];
    }

    // ---- per-block softmax partials in fixed summation order (deterministic) ----
    #pragma unroll
    for (int j = 0; j < 10; ++j) lds_part[wave][lane + 32 * j] = pacc[j];
    __syncthreads();
    for (int e = tid; e < NCODES; e += 256) {
        const int ge = e / V_, ve = e % V_;
        const int wb = ge * 4;
        const float sum = ((lds_part[wb][ve] + lds_part[wb + 1][ve])
                          + lds_part[wb + 2][ve]) + lds_part[wb + 3][ve];
        blockpart[(size_t)blk * NCODES + e] = sum;
    }
}

// ---------------- kernel 2: perplexity scalars (single block, fixed-order) ----------------
__device__ __forceinline__ float block_sum(float v, float* red, int tid) {
    red[tid] = v; __syncthreads();
    for (int s = 128; s > 0; s >>= 1) {
        if (tid < s) red[tid] += red[tid + s];
        __syncthreads();
    }
    const float r = red[0]; __syncthreads();
    return r;
}

__global__ __launch_bounds__(256)
void gvq_final(const int* __restrict__ counts,
               const float* __restrict__ blockpart,
               float* __restrict__ out)
{
    __shared__ float red[256];
    __shared__ float avg_tot[NCODES];
    const int tid = threadIdx.x;

    // total softmax probs over all tokens, fixed block order
    for (int e = tid; e < NCODES; e += 256) {
        float t = 0.0f;
        for (int bk = 0; bk < NBLK; ++bk) t += blockpart[(size_t)bk * NCODES + e];
        avg_tot[e] = t * (1.0f / (float)NTOK);
    }
    __syncthreads();

    // prob perplexity: sum_g exp(-sum_v avg*log(avg+eps))
    float h0 = 0.0f, h1 = 0.0f;
    for (int e = tid; e < NCODES; e += 256) {
        const float a = avg_tot[e];
        const float c = a * logf(a + EPS_);
        if (e < V_) h0 += c; else h1 += c;
    }
    const float H0 = block_sum(h0, red, tid);
    const float H1 = block_sum(h1, red, tid);

    // code perplexity over (T,G) rows of argmax counts / B
    float csum = 0.0f;
    for (int rw = tid; rw < T_ * G_; rw += 256) {
        const int* cp = counts + (size_t)rw * V_;
        float h = 0.0f;
        for (int v = 0; v < V_; ++v) {
            const int c = cp[v];
            if (c) { const float p = (float)c * (1.0f / B_); h += p * logf(p + EPS_); }
        }
        csum += expf(-h);
    }
    const float CP = block_sum(csum, red, tid);

    if (tid == 0) {
        out[QOUT + 0] = CP;                       // code_perplexity
        out[QOUT + 1] = expf(-H0) + expf(-H1);    // prob_perplexity
    }
}

// ---------------- launcher ----------------
extern "C" void kernel_launch(void* const* d_in, const int* in_sizes, int n_in,
                              void* d_out, int out_size, void* d_ws, size_t ws_size,
                              hipStream_t stream) {
    const float* x    = (const float*)d_in[0];   // (B,T,F)
    const float* w    = (const float*)d_in[1];   // (G*V, F)
    const float* bias = (const float*)d_in[2];   // (G*V)
    const float* cb   = (const float*)d_in[3];   // (1, G*V, D)
    const float* gum  = (const float*)d_in[4];   // (B*T, G, V)
    float* out = (float*)d_out;

    int*   counts    = (int*)d_ws;                                   // T*G*V ints
    float* blockpart = (float*)((char*)d_ws +
                        (size_t)T_ * G_ * V_ * sizeof(int));         // NBLK*NCODES floats

    // allow >64KB dynamic LDS for the fused kernel (idempotent, capture-safe)
    (void)hipFuncSetAttribute((const void*)gvq_main,
                              hipFuncAttributeMaxDynamicSharedMemorySize,
                              (int)LDS_BYTES);

    const int nC = T_ * G_ * V_;
    gvq_zero<<<(nC + 255) / 256, 256, 0, stream>>>(counts, nC);
    gvq_main<<<NBLK, 256, (int)LDS_BYTES, stream>>>(x, w, bias, cb, gum, out,
                                                    counts, blockpart);
    gvq_final<<<1, 256, 0, stream>>>(counts, blockpart, out);
}